// LSTM_44547400794593
// MI455X (gfx1250) — compile-verified
//
#include <hip/hip_runtime.h>
#include <hip/hip_bf16.h>
#include <math.h>

#define B_   64
#define S_   1024
#define I_   256
#define H_   512
#define BH   (B_ * H_)        // 32768
#define G4H  (4 * H_)         // 2048

typedef __attribute__((ext_vector_type(2))) float v2f;
typedef __attribute__((ext_vector_type(8))) float v8f;

#define WMMA_F32(a, b, c) \
    __builtin_amdgcn_wmma_f32_16x16x4_f32(false, (a), false, (b), (short)0, (c), false, false)

// ---------------------------------------------------------------------------
// Workspace layout (floats):
//   [0 , 2*BH)          h0buf  (double-buffered layer-0 hidden state)
//   [2*BH , 4*BH)       h1buf  (double-buffered layer-1 hidden state)
//   [4*BH , 5*BH)       c0buf
//   [5*BH , 6*BH)       c1buf
//   [6*BH , 8*BH)       ybuf   (ping-pong layer-0 output -> layer-1 input)
//   [8*BH , 8*BH+4H)    bias0 = b_ih0 + b_hh0
//   [.. +4H , .. +8H)   bias1 = b_ih1 + b_hh1
// ---------------------------------------------------------------------------

__global__ void lstm_init(const float* __restrict__ h0, const float* __restrict__ c0,
                          const float* __restrict__ b_ih0, const float* __restrict__ b_hh0,
                          const float* __restrict__ b_ih1, const float* __restrict__ b_hh1,
                          float* __restrict__ ws) {
    int i = blockIdx.x * blockDim.x + threadIdx.x;
    float* h0buf = ws;
    float* h1buf = ws + 2 * BH;
    float* c0buf = ws + 4 * BH;
    float* c1buf = ws + 5 * BH;
    float* bias0 = ws + 8 * BH;
    float* bias1 = bias0 + G4H;
    if (i < BH) {
        h0buf[i] = h0[i];        // layer 0 initial h  (h0[0])
        h1buf[i] = h0[BH + i];   // layer 1 initial h  (h0[1])
        c0buf[i] = c0[i];
        c1buf[i] = c0[BH + i];
    }
    if (i < G4H) {
        bias0[i] = b_ih0[i] + b_hh0[i];
        bias1[i] = b_ih1[i] + b_hh1[i];
    }
}

// One launch per scan index s. Blocks 0..31: layer 0, step s (if s < S).
// Blocks 32..63: layer 1, step s-1 (if s >= 1). Layer 1 reads the y ping-pong
// buffer written by layer 0 in the PREVIOUS launch -> no intra-kernel dep.
//
// Block = 4 waves, wave = batch tile (mtile). Each wave computes the 16x16
// tiles of ALL FOUR gates for its (mtile, jslice): per K-step it loads one A
// fragment + four B fragments and issues four INDEPENDENT wmma accumulations.
// Since the C/D layout is identical for all four accumulators, gates i,f,g,o
// for a given (m,n) live in the same lane/slot -> pointwise LSTM update runs
// entirely in registers (no LDS, no barrier).
__global__ __launch_bounds__(128)
void lstm_step(int s,
               const float* __restrict__ x,
               const float* __restrict__ Wih0, const float* __restrict__ Whh0,
               const float* __restrict__ Wih1, const float* __restrict__ Whh1,
               float* __restrict__ ws, float* __restrict__ out) {
    const int layer  = blockIdx.x >> 5;   // 0 or 1
    const int jslice = blockIdx.x & 31;   // 16-column slice of H
    if (layer == 0 && s >= S_) return;    // uniform per-block exit (EXEC stays full)
    if (layer == 1 && s == 0)  return;
    const int t = (layer == 0) ? s : (s - 1);

    float* h0buf = ws;
    float* h1buf = ws + 2 * BH;
    float* c0buf = ws + 4 * BH;
    float* c1buf = ws + 5 * BH;
    float* ybuf  = ws + 6 * BH;
    float* bias0 = ws + 8 * BH;
    float* bias1 = bias0 + G4H;

    const float* Wih; const float* Whh; const float* bias;
    const float* hprev; float* hnext; float* cstate;
    const float* xsrc; int xK; long xRowStride;
    if (layer == 0) {
        Wih = Wih0; Whh = Whh0; bias = bias0;
        hprev  = h0buf + (t & 1) * BH;
        hnext  = h0buf + ((t + 1) & 1) * BH;
        cstate = c0buf;
        xsrc = x + (long)t * I_;          // row b lives at b*S*I (+ t*I folded in)
        xK = I_; xRowStride = (long)S_ * I_;
    } else {
        Wih = Wih1; Whh = Whh1; bias = bias1;
        hprev  = h1buf + (t & 1) * BH;
        hnext  = h1buf + ((t + 1) & 1) * BH;
        cstate = c1buf;
        xsrc = ybuf + (t & 1) * BH;       // layer-0 output at step t
        xK = H_; xRowStride = H_;
    }

    const int lane  = threadIdx.x & 31;
    const int mtile = threadIdx.x >> 5;                 // 0..3 (batch tiles)
    const int rowA  = mtile * 16 + (lane & 15);         // batch row 0..63
    const int jcol  = jslice * 16 + (lane & 15);        // column within H
    const int koff  = (lane >> 4) << 1;                 // lanes 0-15: K=k,k+1; 16-31: K=k+2,k+3

    v8f acc0 = {}, acc1 = {}, acc2 = {}, acc3 = {};     // gates i, f, g, o

    // Phase 1: input projection  x_t[64, xK] @ Wih^T  (all 4 gate row-blocks)
    {
        const float* Ap = xsrc + (long)rowA * xRowStride + koff;
        const float* B0 = Wih + (long)(0 * H_ + jcol) * xK + koff;
        const float* B1 = Wih + (long)(1 * H_ + jcol) * xK + koff;
        const float* B2 = Wih + (long)(2 * H_ + jcol) * xK + koff;
        const float* B3 = Wih + (long)(3 * H_ + jcol) * xK + koff;
        #pragma unroll 2
        for (int k = 0; k < xK; k += 4) {
            v2f a  = *(const v2f*)(Ap + k);
            v2f b0 = *(const v2f*)(B0 + k);
            v2f b1 = *(const v2f*)(B1 + k);
            v2f b2 = *(const v2f*)(B2 + k);
            v2f b3 = *(const v2f*)(B3 + k);
            acc0 = WMMA_F32(a, b0, acc0);
            acc1 = WMMA_F32(a, b1, acc1);
            acc2 = WMMA_F32(a, b2, acc2);
            acc3 = WMMA_F32(a, b3, acc3);
        }
    }
    // Phase 2: recurrent projection  h_t[64, H] @ Whh^T
    {
        const float* Ap = hprev + (long)rowA * H_ + koff;
        const float* B0 = Whh + (long)(0 * H_ + jcol) * H_ + koff;
        const float* B1 = Whh + (long)(1 * H_ + jcol) * H_ + koff;
        const float* B2 = Whh + (long)(2 * H_ + jcol) * H_ + koff;
        const float* B3 = Whh + (long)(3 * H_ + jcol) * H_ + koff;
        #pragma unroll 2
        for (int k = 0; k < H_; k += 4) {
            v2f a  = *(const v2f*)(Ap + k);
            v2f b0 = *(const v2f*)(B0 + k);
            v2f b1 = *(const v2f*)(B1 + k);
            v2f b2 = *(const v2f*)(B2 + k);
            v2f b3 = *(const v2f*)(B3 + k);
            acc0 = WMMA_F32(a, b0, acc0);
            acc1 = WMMA_F32(a, b1, acc1);
            acc2 = WMMA_F32(a, b2, acc2);
            acc3 = WMMA_F32(a, b3, acc3);
        }
    }

    // In-register pointwise LSTM update. C/D layout: slot r, lane l holds
    // (m = r + 8*(l/16), n = l%16) -- identical across the four accumulators,
    // so i,f,g,o for each element are already co-resident in this lane.
    const int n = lane & 15;
    const int j = jslice * 16 + n;
    const float bi = bias[j];
    const float bf = bias[H_ + j];
    const float bg = bias[2 * H_ + j];
    const float bo = bias[3 * H_ + j];
    const int mbase = mtile * 16 + 8 * (lane >> 4);
    #pragma unroll
    for (int r = 0; r < 8; ++r) {
        const int bb = mbase + r;                 // batch row
        const long idx = (long)bb * H_ + j;
        float gi = acc0[r] + bi;
        float gf = acc1[r] + bf;
        float gg = acc2[r] + bg;
        float go = acc3[r] + bo;
        float si = 1.0f / (1.0f + expf(-gi));
        float sf = 1.0f / (1.0f + expf(-gf));
        float so = 1.0f / (1.0f + expf(-go));
        float tg = tanhf(gg);
        float cnew = sf * cstate[idx] + si * tg;
        float hh   = so * tanhf(cnew);
        cstate[idx] = cnew;
        hnext[idx]  = hh;
        if (layer == 0) {
            ybuf[(t & 1) * BH + idx] = hh;            // feed layer 1 next launch
        } else {
            out[((long)bb * S_ + t) * H_ + j] = hh;   // [B, S, H]
        }
    }
}

__global__ void lstm_final(const float* __restrict__ ws, float* __restrict__ out) {
    int i = blockIdx.x * blockDim.x + threadIdx.x;
    if (i >= BH) return;
    const float* h0buf = ws;
    const float* h1buf = ws + 2 * BH;
    const float* c0buf = ws + 4 * BH;
    const float* c1buf = ws + 5 * BH;
    const long O = (long)B_ * S_ * H_;
    out[O + 0 * BH + i] = h0buf[(S_ & 1) * BH + i];  // final h, layer 0
    out[O + 1 * BH + i] = h1buf[(S_ & 1) * BH + i];  // final h, layer 1
    out[O + 2 * BH + i] = c0buf[i];                  // final c, layer 0
    out[O + 3 * BH + i] = c1buf[i];                  // final c, layer 1
}

extern "C" void kernel_launch(void* const* d_in, const int* in_sizes, int n_in,
                              void* d_out, int out_size, void* d_ws, size_t ws_size,
                              hipStream_t stream) {
    const float* x    = (const float*)d_in[0];
    const float* h0   = (const float*)d_in[1];
    const float* c0   = (const float*)d_in[2];
    const float* Wih0 = (const float*)d_in[3];
    const float* Whh0 = (const float*)d_in[4];
    const float* bih0 = (const float*)d_in[5];
    const float* bhh0 = (const float*)d_in[6];
    const float* Wih1 = (const float*)d_in[7];
    const float* Whh1 = (const float*)d_in[8];
    const float* bih1 = (const float*)d_in[9];
    const float* bhh1 = (const float*)d_in[10];
    float* out = (float*)d_out;
    float* ws  = (float*)d_ws;

    lstm_init<<<(BH + 255) / 256, 256, 0, stream>>>(h0, c0, bih0, bhh0, bih1, bhh1, ws);
    for (int s = 0; s <= S_; ++s) {
        lstm_step<<<64, 128, 0, stream>>>(s, x, Wih0, Whh0, Wih1, Whh1, ws, out);
    }
    lstm_final<<<(BH + 255) / 256, 256, 0, stream>>>(ws, out);
}